// DecoderRNN_40140764348627
// MI455X (gfx1250) — compile-verified
//
#include <hip/hip_runtime.h>
#include <math.h>
#include <stdint.h>

// Problem dims (fixed by reference)
#define BB     128      // batch
#define SS     40       // encoder seq len
#define HH     512      // hidden
#define WD     512      // word dim
#define VV     32000    // vocab
#define TSTEPS 27       // MAXLEN-1
#define LDT    28       // targets row stride

typedef float v2f __attribute__((ext_vector_type(2)));
typedef float v8f __attribute__((ext_vector_type(8)));
typedef int   v4i_ __attribute__((ext_vector_type(4)));

__device__ __forceinline__ float sigmoidf_(float x) { return 1.0f / (1.0f + __expf(-x)); }

// CDNA5 async global->LDS path (guarded; falls back to load+ds_store)
#if defined(__AMDGCN__) && __has_builtin(__builtin_amdgcn_global_load_async_to_lds_b128) && \
    __has_builtin(__builtin_amdgcn_s_wait_asynccnt)
#define USE_ASYNC_LDS 1
typedef __attribute__((address_space(1))) v4i_ gv4i_t;   // global (clang prints as __device__)
typedef __attribute__((address_space(3))) v4i_ lv4i_t;   // LDS
#else
#define USE_ASYNC_LDS 0
#endif

// ---------------------------------------------------------------------------
// fp32 WMMA GEMM:  C[m][n] = sum_k A[m*lda+k] * W[n*ldb+kofs+k] (+bias[n])
//
// Block = 256 threads = 8 waves; workgroup tile = 128(M) x 64(N):
//   wave w owns M rows [16w, 16w+16) with four 16x16 accumulators along N.
// W tile (64 x 32) is staged in LDS (double buffered, async prefetch) so each
// W element is fetched from L2 once per workgroup instead of once per 16-row
// M tile.  Requires: M % 128 == 0, N % 64 == 0, K % 32 == 0 (all true here).
// ---------------------------------------------------------------------------
#define KB    32                 // K block
#define NB    64                 // N block
#define BROW  (KB + 4)           // LDS row pitch (floats): 36 -> conflict-free
__global__ __launch_bounds__(256)
void gemm_wmma_f32_kernel(const float* __restrict__ A, const float* __restrict__ W,
                          const float* __restrict__ bias, float* __restrict__ C,
                          int K, int lda, int ldb, int kofs, int ldc)
{
    __shared__ float Bs[2][NB * BROW];

    const int tid  = threadIdx.x;
    const int lane = tid & 31;
    const int wave = tid >> 5;                 // 0..7 -> M chunk
    const int m0   = blockIdx.y * 128;
    const int n0   = blockIdx.x * NB;
    const int l15  = lane & 15;
    const int klo  = (lane >> 4) << 1;         // lanes 0-15 -> K{0,1}; 16-31 -> K{2,3}

    const float* Ap = A + (size_t)(m0 + wave * 16 + l15) * lda + klo;

    // Cooperative staging map: thread -> rows {tid>>3, tid>>3 + 32}, 16B segment tid&7
    const int sn   = tid >> 3;                 // 0..31
    const int sseg = tid & 7;                  // 0..7  (8 x float4 per 32-float row)

    auto stage = [&](int buf, int k0) {
        #pragma unroll
        for (int h = 0; h < 2; ++h) {
            const int n = sn + 32 * h;
            const float* src = W + (size_t)(n0 + n) * ldb + kofs + k0 + sseg * 4;
            float* dst = &Bs[buf][n * BROW + sseg * 4];
#if USE_ASYNC_LDS
            __builtin_amdgcn_global_load_async_to_lds_b128(
                (gv4i_t*)(uintptr_t)src,
                (lv4i_t*)(uint32_t)(uintptr_t)dst, 0, 0);
#else
            *(float4*)dst = *(const float4*)src;
#endif
        }
    };

    v8f acc[4];
    acc[0] = (v8f){}; acc[1] = (v8f){}; acc[2] = (v8f){}; acc[3] = (v8f){};

    const int nkb = K / KB;
    stage(0, 0);
    int buf = 0;

    for (int kb = 0; kb < nkb; ++kb) {
        const int k0 = kb * KB;
        const bool more = (kb + 1 < nkb);
        if (more) stage(buf ^ 1, k0 + KB);     // prefetch next tile into other buffer
#if USE_ASYNC_LDS
        if (more) __builtin_amdgcn_s_wait_asynccnt(2);   // current tile's 2 xfers done
        else      __builtin_amdgcn_s_wait_asynccnt(0);
#endif
        __syncthreads();                       // all waves' tile writes visible

        // Batch-load this k-block's A fragments (hot in WGP$/L2)
        v2f a[KB / 4];
        #pragma unroll
        for (int kk = 0; kk < KB / 4; ++kk) {
            a[kk].x = Ap[k0 + kk * 4];
            a[kk].y = Ap[k0 + kk * 4 + 1];
        }

        const float* bsrc = &Bs[buf][l15 * BROW + klo];
        #pragma unroll
        for (int kk = 0; kk < KB / 4; ++kk) {
            v2f b0 = *(const v2f*)(bsrc + 0 * 16 * BROW + kk * 4);
            v2f b1 = *(const v2f*)(bsrc + 1 * 16 * BROW + kk * 4);
            v2f b2 = *(const v2f*)(bsrc + 2 * 16 * BROW + kk * 4);
            v2f b3 = *(const v2f*)(bsrc + 3 * 16 * BROW + kk * 4);
            acc[0] = __builtin_amdgcn_wmma_f32_16x16x4_f32(false, a[kk], false, b0, (short)0, acc[0], false, false);
            acc[1] = __builtin_amdgcn_wmma_f32_16x16x4_f32(false, a[kk], false, b1, (short)0, acc[1], false, false);
            acc[2] = __builtin_amdgcn_wmma_f32_16x16x4_f32(false, a[kk], false, b2, (short)0, acc[2], false, false);
            acc[3] = __builtin_amdgcn_wmma_f32_16x16x4_f32(false, a[kk], false, b3, (short)0, acc[3], false, false);
        }
        __syncthreads();                       // tile fully consumed before overwrite
        buf ^= 1;
    }

    // C/D layout: element(vgpr i, lane) = C[mrow + i][n0 + 16t + (lane&15)]
    const int mb = m0 + wave * 16 + ((lane >> 4) << 3);
    #pragma unroll
    for (int t = 0; t < 4; ++t) {
        const int nn = n0 + t * 16 + l15;
        const float bv = bias ? bias[nn] : 0.0f;
        #pragma unroll
        for (int i = 0; i < 8; ++i)
            C[(size_t)(mb + i) * ldc + nn] = acc[t][i] + bv;
    }
}

// ---------------------------------------------------------------------------
__global__ __launch_bounds__(256)
void zero_kernel(float* __restrict__ p, int n)
{
    int i = blockIdx.x * 256 + threadIdx.x;
    if (i < n) p[i] = 0.0f;
}

// mean over S of encoder_outputs -> (B, H)
__global__ __launch_bounds__(256)
void mean_ctx_kernel(const float* __restrict__ enc, float* __restrict__ mctx)
{
    int idx = blockIdx.x * 256 + threadIdx.x;          // < B*H
    int b = idx >> 9, j = idx & (HH - 1);
    const float* e = enc + (size_t)b * SS * HH + j;
    float s = 0.0f;
    for (int k = 0; k < SS; ++k) s += e[(size_t)k * HH];
    mctx[idx] = s * (1.0f / SS);
}

// xcat[b] = [ embedding[targets[b][t]] ; ctx[b] ]
__global__ __launch_bounds__(256)
void build_x_kernel(const float* __restrict__ emb, const int* __restrict__ tgt, int t,
                    const float* __restrict__ ctx, float* __restrict__ x)
{
    int idx = blockIdx.x * 256 + threadIdx.x;          // < B*(WD+HH)
    int b = idx >> 10, j = idx & 1023;
    float v;
    if (j < WD) v = emb[(size_t)tgt[b * LDT + t] * WD + j];
    else        v = ctx[(size_t)b * HH + (j - WD)];
    x[idx] = v;
}

// e[s] = w2 . tanh(enc_proj[b,s,:] + hid_proj[b,:]); alpha = softmax(e);
// ctx[b,:] = sum_s alpha[s] * enc[b,s,:].  One workgroup per batch row.
__global__ __launch_bounds__(256)
void attn_kernel(const float* __restrict__ enc, const float* __restrict__ enc_proj,
                 const float* __restrict__ hid_proj, const float* __restrict__ w2,
                 float* __restrict__ ctx)
{
    const int b = blockIdx.x;
    const int tid = threadIdx.x;
    const int lane = tid & 31, wave = tid >> 5;
    __shared__ float sE[SS];
    __shared__ float sA[SS];

    const float* ep = enc_proj + (size_t)b * SS * HH;
    const float* hp = hid_proj + (size_t)b * HH;

    for (int s = wave; s < SS; s += 8) {
        float p = 0.0f;
        const float* eps = ep + (size_t)s * HH;
        for (int j = lane; j < HH; j += 32)
            p += w2[j] * tanhf(eps[j] + hp[j]);
        #pragma unroll
        for (int off = 16; off; off >>= 1) p += __shfl_down(p, off, 32);
        if (lane == 0) sE[s] = p;
    }
    __syncthreads();

    if (wave == 0) {
        float x0 = (lane < SS) ? sE[lane] : -INFINITY;
        float x1 = (lane + 32 < SS) ? sE[lane + 32] : -INFINITY;
        float m = fmaxf(x0, x1);
        #pragma unroll
        for (int off = 16; off; off >>= 1) m = fmaxf(m, __shfl_xor(m, off, 32));
        float e0 = (lane < SS) ? __expf(x0 - m) : 0.0f;
        float e1 = (lane + 32 < SS) ? __expf(x1 - m) : 0.0f;
        float ssum = e0 + e1;
        #pragma unroll
        for (int off = 16; off; off >>= 1) ssum += __shfl_xor(ssum, off, 32);
        float inv = 1.0f / ssum;
        if (lane < SS) sA[lane] = e0 * inv;
        if (lane + 32 < SS) sA[lane + 32] = e1 * inv;
    }
    __syncthreads();

    const float* eb = enc + (size_t)b * SS * HH;
    for (int j = tid; j < HH; j += 256) {
        float acc = 0.0f;
        for (int s = 0; s < SS; ++s) acc += sA[s] * eb[(size_t)s * HH + j];
        ctx[(size_t)b * HH + j] = acc;
    }
}

// GRU gate nonlinearity:  hid_new = (1-z)*n + z*hid
__global__ __launch_bounds__(256)
void gru_gate_kernel(const float* __restrict__ gi, const float* __restrict__ gh,
                     const float* __restrict__ hid, float* __restrict__ hid_new)
{
    int idx = blockIdx.x * 256 + threadIdx.x;          // < B*H
    int b = idx >> 9, j = idx & (HH - 1);
    const float* gib = gi + (size_t)b * 3 * HH;
    const float* ghb = gh + (size_t)b * 3 * HH;
    float r = sigmoidf_(gib[j] + ghb[j]);
    float z = sigmoidf_(gib[HH + j] + ghb[HH + j]);
    float n = tanhf(gib[2 * HH + j] + r * ghb[2 * HH + j]);
    hid_new[idx] = (1.0f - z) * n + z * hid[idx];
}

// In-place log_softmax over one vocab row per workgroup.
__global__ __launch_bounds__(256)
void logsoftmax_kernel(float* __restrict__ out, int t)
{
    const int b = blockIdx.x;
    float* row = out + ((size_t)b * TSTEPS + t) * VV;
    const int tid = threadIdx.x, lane = tid & 31, wave = tid >> 5;
    __shared__ float red[8];
    __shared__ float sM, sL;

    float m = -INFINITY;
    for (int j = tid; j < VV; j += 256) m = fmaxf(m, row[j]);
    #pragma unroll
    for (int off = 16; off; off >>= 1) m = fmaxf(m, __shfl_xor(m, off, 32));
    if (lane == 0) red[wave] = m;
    __syncthreads();
    if (tid == 0) {
        float mm = red[0];
        for (int i = 1; i < 8; ++i) mm = fmaxf(mm, red[i]);
        sM = mm;
    }
    __syncthreads();
    const float M = sM;
    __syncthreads();

    float s = 0.0f;
    for (int j = tid; j < VV; j += 256) s += __expf(row[j] - M);
    #pragma unroll
    for (int off = 16; off; off >>= 1) s += __shfl_xor(s, off, 32);
    if (lane == 0) red[wave] = s;
    __syncthreads();
    if (tid == 0) {
        float tt = 0.0f;
        for (int i = 0; i < 8; ++i) tt += red[i];
        sL = M + __logf(tt);
    }
    __syncthreads();
    const float L = sL;
    for (int j = tid; j < VV; j += 256) row[j] -= L;
}

// ---------------------------------------------------------------------------
extern "C" void kernel_launch(void* const* d_in, const int* in_sizes, int n_in,
                              void* d_out, int out_size, void* d_ws, size_t ws_size,
                              hipStream_t stream)
{
    (void)in_sizes; (void)n_in; (void)out_size; (void)ws_size;

    const float* enc   = (const float*)d_in[0];   // (B,S,H)
    // d_in[1] = encoder_hidden: dead code in reference
    const float* emb   = (const float*)d_in[2];   // (V,WD)
    const float* W_ih  = (const float*)d_in[3];   // (3H, H+WD)
    const float* W_hh  = (const float*)d_in[4];   // (3H, H)
    const float* b_ih  = (const float*)d_in[5];
    const float* b_hh  = (const float*)d_in[6];
    const float* aW1   = (const float*)d_in[7];   // (H, 2H)
    const float* ab1   = (const float*)d_in[8];
    const float* aw2   = (const float*)d_in[9];
    const float* out_W = (const float*)d_in[10];  // (V, H)
    const float* out_b = (const float*)d_in[11];
    const int*   tgt   = (const int*)d_in[12];    // (B, 28)
    float* out = (float*)d_out;                   // (B, 27, V)

    float* ws = (float*)d_ws;
    float* enc_proj = ws;  ws += (size_t)BB * SS * HH;   // enc @ W1e^T + b1 (pre-tanh)
    float* mctx     = ws;  ws += BB * HH;
    float* hidA     = ws;  ws += BB * HH;
    float* hidB     = ws;  ws += BB * HH;
    float* hproj    = ws;  ws += BB * HH;
    float* ctx      = ws;  ws += BB * HH;
    float* xcat     = ws;  ws += BB * (WD + HH);
    float* gi       = ws;  ws += BB * 3 * HH;
    float* gh       = ws;  /* last */

    // h0 = 0 (ws is poisoned by harness)
    zero_kernel<<<(BB * HH) / 256, 256, 0, stream>>>(hidA, BB * HH);
    // step-0 context = mean over frames
    mean_ctx_kernel<<<(BB * HH) / 256, 256, 0, stream>>>(enc, mctx);
    // hoisted attention encoder projection: (B*S,H) @ W1e^T + b1
    gemm_wmma_f32_kernel<<<dim3(HH / NB, (BB * SS) / 128), 256, 0, stream>>>(
        enc, aW1, ab1, enc_proj, HH, HH, 2 * HH, 0, HH);

    float* hid = hidA;
    float* hid_new = hidB;

    for (int t = 0; t < TSTEPS; ++t) {
        const float* ctx_t;
        if (t == 0) {
            ctx_t = mctx;
        } else {
            // hid projection through W1h = attn_W1[:, H:2H]  (no bias; b1 is in enc_proj)
            gemm_wmma_f32_kernel<<<dim3(HH / NB, 1), 256, 0, stream>>>(
                hid, aW1, nullptr, hproj, HH, HH, 2 * HH, HH, HH);
            attn_kernel<<<BB, 256, 0, stream>>>(enc, enc_proj, hproj, aw2, ctx);
            ctx_t = ctx;
        }

        build_x_kernel<<<(BB * (WD + HH)) / 256, 256, 0, stream>>>(emb, tgt, t, ctx_t, xcat);

        // gi = xcat @ W_ih^T + b_ih   (128 x 1536, K=1024)
        gemm_wmma_f32_kernel<<<dim3((3 * HH) / NB, 1), 256, 0, stream>>>(
            xcat, W_ih, b_ih, gi, WD + HH, WD + HH, WD + HH, 0, 3 * HH);
        // gh = hid @ W_hh^T + b_hh    (128 x 1536, K=512)
        gemm_wmma_f32_kernel<<<dim3((3 * HH) / NB, 1), 256, 0, stream>>>(
            hid, W_hh, b_hh, gh, HH, HH, HH, 0, 3 * HH);

        gru_gate_kernel<<<(BB * HH) / 256, 256, 0, stream>>>(gi, gh, hid, hid_new);

        // logits straight into d_out[b][t][:]  (128 x 32000, K=512); ldc = 27*V
        gemm_wmma_f32_kernel<<<dim3(VV / NB, 1), 256, 0, stream>>>(
            hid_new, out_W, out_b, out + (size_t)t * VV, HH, HH, HH, 0, TSTEPS * VV);

        logsoftmax_kernel<<<BB, 256, 0, stream>>>(out, t);

        float* tmp = hid; hid = hid_new; hid_new = tmp;
    }
}